// SAGE_86466281603835
// MI455X (gfx1250) — compile-verified
//
#include <hip/hip_runtime.h>
#include <hip/hip_bf16.h>

typedef float v2f __attribute__((ext_vector_type(2)));
typedef float v8f __attribute__((ext_vector_type(8)));

// ---------------------------------------------------------------------------
// Zero a float buffer (grid-stride).
// ---------------------------------------------------------------------------
__global__ void sage_zero_f32(float* __restrict__ p, long long n) {
    long long i = (long long)blockIdx.x * blockDim.x + threadIdx.x;
    long long stride = (long long)gridDim.x * blockDim.x;
    for (; i < n; i += stride) p[i] = 0.0f;
}

// ---------------------------------------------------------------------------
// Edge scatter: one wave (32 lanes) per edge.
//   agg[dst][:] += feat[src][:]   (d = 128 or 256, float atomics -> L2)
//   deg[dst]    += 1              (lane 0, only when deg != nullptr)
// Gather is float4 per lane (512B coalesced per wave per 128 cols).
// feat (51-102 MB) is L2-resident (192 MB L2), so the random gather/scatter
// stays on-chip.
// ---------------------------------------------------------------------------
__global__ void sage_scatter(const float* __restrict__ feat,
                             const long long* __restrict__ ei,
                             int E, int d,
                             float* __restrict__ agg,
                             float* __restrict__ deg) {
    int gw   = (int)(((long long)blockIdx.x * blockDim.x + threadIdx.x) >> 5);
    int lane = threadIdx.x & 31;
    if (gw >= E) return;

    long long s = ei[gw];                 // edge_index[0][gw]  (src)
    long long t = ei[(long long)E + gw];  // edge_index[1][gw]  (dst)

    const float4* xs = (const float4*)(feat + s * (long long)d);
    float*        ap = agg + t * (long long)d;

    int chunks = d >> 7;  // 128 floats (= 32 lanes * float4) per chunk
    for (int i = 0; i < chunks; ++i) {
        int idx = i * 32 + lane;          // float4 index
        float4 v = xs[idx];
        float* a = ap + idx * 4;
        atomicAdd(a + 0, v.x);
        atomicAdd(a + 1, v.y);
        atomicAdd(a + 2, v.z);
        atomicAdd(a + 3, v.w);
    }
    if (deg != nullptr && lane == 0) atomicAdd(deg + t, 1.0f);
}

// ---------------------------------------------------------------------------
// Fused SAGE transform via fp32 WMMA:
//   out[i,:] = act( (agg[i,:]/max(deg[i],1)) @ Wl + bias + xin[i,:] @ Wr )
//
// One wave computes a 32-row x 64-col output tile (2 row tiles x 4 col tiles,
// acc[2][4] = 64 accumulator VGPRs). Each B (weight) fragment load feeds two
// WMMAs -> 16 WMMAs per 16 B-loads per k-step, halving weight re-read traffic
// vs one row tile per wave.
//   grid.x  = N/32 row-tile pairs
//   block.x = 32 * (OUT/64)  -> wave id selects the 64-col group
//
// V_WMMA_F32_16X16X4_F32 fragment layout (ISA 7.12.2):
//   A (16x4):  lane m=lane&15, half=lane>>4; holds A[m][2*half + {0,1}]
//   B (4x16):  lane holds B[2*half + {0,1}][m]
//   C/D v8f:   elem r holds D[r + 8*half][m]
// ---------------------------------------------------------------------------
__global__ void sage_wmma_gemm(const float* __restrict__ agg,
                               const float* __restrict__ deg,
                               const float* __restrict__ xin,
                               const float* __restrict__ Wl,
                               const float* __restrict__ bias,
                               const float* __restrict__ Wr,
                               float* __restrict__ out,
                               int K, int OUT, int do_relu) {
    const int lane = threadIdx.x & 31;
    const int wave = threadIdx.x >> 5;
    const int m    = lane & 15;
    const int half = lane >> 4;           // 0: K offset 0..1, 1: K offset 2..3
    const int row0 = blockIdx.x * 32;     // two 16-row tiles per block
    const int rA   = row0 + m;
    const int rB   = row0 + 16 + m;
    const int c0   = wave * 64;

    const float invdA = 1.0f / fmaxf(deg[rA], 1.0f);
    const float invdB = 1.0f / fmaxf(deg[rB], 1.0f);
    const float* aggA = agg + (size_t)rA * K;
    const float* aggB = agg + (size_t)rB * K;
    const float* xA   = xin + (size_t)rA * K;
    const float* xB   = xin + (size_t)rB * K;

    v8f acc[2][4];
#pragma unroll
    for (int rt = 0; rt < 2; ++rt)
#pragma unroll
        for (int t = 0; t < 4; ++t)
            acc[rt][t] = (v8f){0.f,0.f,0.f,0.f,0.f,0.f,0.f,0.f};

    for (int k = 0; k < K; k += 4) {
        const int ka = k + 2 * half;

        v2f a0, a1;   // mean fragments for the two row tiles
        a0.x = aggA[ka] * invdA;  a0.y = aggA[ka + 1] * invdA;
        a1.x = aggB[ka] * invdB;  a1.y = aggB[ka + 1] * invdB;
        v2f x0, x1;   // self fragments
        x0.x = xA[ka];  x0.y = xA[ka + 1];
        x1.x = xB[ka];  x1.y = xB[ka + 1];

#pragma unroll
        for (int t = 0; t < 4; ++t) {
            const int c = c0 + t * 16 + m;
            v2f bl;
            bl.x = Wl[(size_t)ka * OUT + c];
            bl.y = Wl[(size_t)(ka + 1) * OUT + c];
            acc[0][t] = __builtin_amdgcn_wmma_f32_16x16x4_f32(
                false, a0, false, bl, (short)0, acc[0][t], false, false);
            acc[1][t] = __builtin_amdgcn_wmma_f32_16x16x4_f32(
                false, a1, false, bl, (short)0, acc[1][t], false, false);
            v2f br;
            br.x = Wr[(size_t)ka * OUT + c];
            br.y = Wr[(size_t)(ka + 1) * OUT + c];
            acc[0][t] = __builtin_amdgcn_wmma_f32_16x16x4_f32(
                false, x0, false, br, (short)0, acc[0][t], false, false);
            acc[1][t] = __builtin_amdgcn_wmma_f32_16x16x4_f32(
                false, x1, false, br, (short)0, acc[1][t], false, false);
        }
    }

    // Epilogue: bias (+ReLU) and store. D elem r -> row (r + 8*half), col m.
#pragma unroll
    for (int rt = 0; rt < 2; ++rt) {
#pragma unroll
        for (int t = 0; t < 4; ++t) {
            const int c  = c0 + t * 16 + m;
            const float bv = bias[c];
#pragma unroll
            for (int r = 0; r < 8; ++r) {
                const int orow = row0 + rt * 16 + r + 8 * half;
                float v = acc[rt][t][r] + bv;
                if (do_relu) v = fmaxf(v, 0.0f);
                out[(size_t)orow * OUT + c] = v;
            }
        }
    }
}

// ---------------------------------------------------------------------------
// Host launcher. Inputs (setup_inputs order):
//  0: x [N,128] f32          1: edge_index [2,E] int64
//  2: W_l1 [128,256]  3: b_l1 [256]  4: W_r1 [128,256]
//  5: W_l2 [256,128]  6: b_l2 [128]  7: W_r2 [256,128]
// Output: [N,128] f32.
// Workspace: deg[N] | h[N*256] | buf[N*256]  (~205 MB)
// ---------------------------------------------------------------------------
extern "C" void kernel_launch(void* const* d_in, const int* in_sizes, int n_in,
                              void* d_out, int out_size, void* d_ws, size_t ws_size,
                              hipStream_t stream) {
    const float*     x   = (const float*)d_in[0];
    const long long* ei  = (const long long*)d_in[1];
    const float*     Wl1 = (const float*)d_in[2];
    const float*     bl1 = (const float*)d_in[3];
    const float*     Wr1 = (const float*)d_in[4];
    const float*     Wl2 = (const float*)d_in[5];
    const float*     bl2 = (const float*)d_in[6];
    const float*     Wr2 = (const float*)d_in[7];
    float*           out = (float*)d_out;

    const int N = in_sizes[0] / 128;   // 100000 (divisible by 32)
    const int E = in_sizes[1] / 2;     // 1600000

    float* deg = (float*)d_ws;
    float* h   = deg + N;
    float* buf = h + (size_t)N * 256;

    const int scatterBlocks = (E + 7) / 8;  // 8 waves (edges) per 256-thr block

    // --- Layer 1 ---
    sage_zero_f32<<<1024, 256, 0, stream>>>(deg, (long long)N);
    sage_zero_f32<<<8192, 256, 0, stream>>>(buf, (long long)N * 128);
    sage_scatter<<<scatterBlocks, 256, 0, stream>>>(x, ei, E, 128, buf, deg);
    sage_wmma_gemm<<<N / 32, 128, 0, stream>>>(buf, deg, x, Wl1, bl1, Wr1, h,
                                               128, 256, 1);

    // --- Layer 2 (deg identical: same edge list) ---
    sage_zero_f32<<<8192, 256, 0, stream>>>(buf, (long long)N * 256);
    sage_scatter<<<scatterBlocks, 256, 0, stream>>>(h, ei, E, 256, buf, nullptr);
    sage_wmma_gemm<<<N / 32, 64, 0, stream>>>(buf, deg, h, Wl2, bl2, Wr2, out,
                                              256, 128, 0);
}